// GeometricConstraints_60876866453862
// MI455X (gfx1250) — compile-verified
//
#include <hip/hip_runtime.h>
#include <math.h>

#define NPTS 8192
#define EPSL 1e-8f
#define RAD2 0.0025f   // 0.05^2

typedef __attribute__((ext_vector_type(2))) float v2f;
typedef __attribute__((ext_vector_type(8))) float v8f;

// Workspace layout (float slots):
//  0 sum_pred, 1 sum_gt, 2 n_head, 3 hx, 4 hy, 5 hz, 6 sum_bf,
//  7 Sd (sum hf*d), 8 Sd2 (sum hf*d^2), 9 maxd (float bits, uint atomicMax),
// 10..15 cov sums (xx,xy,xz,yy,yz,zz), 16 sum_var, 17 ni

__device__ __forceinline__ float wred_sum(float v) {
  #pragma unroll
  for (int m = 1; m < 32; m <<= 1) v += __shfl_xor(v, m, 32);
  return v;
}
__device__ __forceinline__ float wred_max(float v) {
  #pragma unroll
  for (int m = 1; m < 32; m <<= 1) v = fmaxf(v, __shfl_xor(v, m, 32));
  return v;
}

__global__ void k_init(float* ws) {
  if (threadIdx.x < 32) ws[threadIdx.x] = 0.0f;  // slot 9 = 0.0f bits, valid for uint max of d>=0
}

__global__ void k_reduce1(const float* __restrict__ pts,
                          const int* __restrict__ hm,
                          const int* __restrict__ pm,
                          const int* __restrict__ gm,
                          const int* __restrict__ bm,
                          float* ws) {
  float sp = 0.f, sg = 0.f, sb = 0.f, nh = 0.f, hx = 0.f, hy = 0.f, hz = 0.f;
  for (int i = blockIdx.x * blockDim.x + threadIdx.x; i < NPTS;
       i += gridDim.x * blockDim.x) {
    if (pm[i]) sp += 1.f;
    if (gm[i]) sg += 1.f;
    if (bm[i]) sb += 1.f;
    if (hm[i]) {
      nh += 1.f;
      hx += pts[i * 3 + 0];
      hy += pts[i * 3 + 1];
      hz += pts[i * 3 + 2];
    }
  }
  sp = wred_sum(sp); sg = wred_sum(sg); sb = wred_sum(sb); nh = wred_sum(nh);
  hx = wred_sum(hx); hy = wred_sum(hy); hz = wred_sum(hz);
  if ((threadIdx.x & 31) == 0) {
    atomicAdd(&ws[0], sp); atomicAdd(&ws[1], sg); atomicAdd(&ws[2], nh);
    atomicAdd(&ws[3], hx); atomicAdd(&ws[4], hy); atomicAdd(&ws[5], hz);
    atomicAdd(&ws[6], sb);
  }
}

__global__ void k_reduce2(const float* __restrict__ pts,
                          const int* __restrict__ hm,
                          float* ws) {
  const float n  = ws[2];
  const float cx = ws[3] / n, cy = ws[4] / n, cz = ws[5] / n;
  float Sd = 0.f, Sd2 = 0.f, mx = 0.f;
  float xx = 0.f, xy = 0.f, xz = 0.f, yy = 0.f, yz = 0.f, zz = 0.f;
  for (int i = blockIdx.x * blockDim.x + threadIdx.x; i < NPTS;
       i += gridDim.x * blockDim.x) {
    if (hm[i]) {
      float dx = pts[i * 3 + 0] - cx;
      float dy = pts[i * 3 + 1] - cy;
      float dz = pts[i * 3 + 2] - cz;
      float d2 = dx * dx + dy * dy + dz * dz;
      float d  = sqrtf(d2);
      Sd += d; Sd2 += d2; mx = fmaxf(mx, d);
      xx += dx * dx; xy += dx * dy; xz += dx * dz;
      yy += dy * dy; yz += dy * dz; zz += dz * dz;
    }
  }
  Sd = wred_sum(Sd); Sd2 = wred_sum(Sd2); mx = wred_max(mx);
  xx = wred_sum(xx); xy = wred_sum(xy); xz = wred_sum(xz);
  yy = wred_sum(yy); yz = wred_sum(yz); zz = wred_sum(zz);
  if ((threadIdx.x & 31) == 0) {
    atomicAdd(&ws[7], Sd);  atomicAdd(&ws[8], Sd2);
    atomicMax((unsigned*)ws + 9, __float_as_uint(mx));
    atomicAdd(&ws[10], xx); atomicAdd(&ws[11], xy); atomicAdd(&ws[12], xz);
    atomicAdd(&ws[13], yy); atomicAdd(&ws[14], yz); atomicAdd(&ws[15], zz);
  }
}

// One wave = one 16-row tile; sweeps all 512 column tiles with
// V_WMMA_F32_16X16X4_F32 computing the 16x16 block of P@P^T (K=3 padded to 4).
//
// Neighbor test folded into a single compare:
//   d2 < R2  <=>  dot > 0.5*sq_i + 0.5*sq_j - 0.5*R2
// rowc[r] = 0.5*sq_i - 0.5*R2 is loop-invariant; hs = bf_j ? 0.5*sq_j : +inf
// folds the column boundary mask (inf makes every row compare fail).
// The row boundary mask bf_i only gates the final include, so it is applied
// in the epilogue, not in the inner loop.
__global__ void __launch_bounds__(256) k_smooth(const float* __restrict__ pts,
                                                const float* __restrict__ seg,
                                                const int* __restrict__ bm,
                                                float* ws) {
  const int lane = threadIdx.x & 31;
  const int wv   = threadIdx.x >> 5;          // 0..7
  const int rt   = blockIdx.x * 8 + wv;       // row tile 0..511
  const int l    = lane & 15;
  const bool hi  = lane >= 16;

  // A operand (16x4 f32): lanes 0-15 -> (K0,K1)=(x,y); lanes 16-31 -> (K2,K3)=(z,0)
  const int arow = rt * 16 + l;
  const float ax = pts[arow * 3 + 0];
  const float ay = pts[arow * 3 + 1];
  const float az = pts[arow * 3 + 2];
  v2f A;
  A.x = hi ? az : ax;
  A.y = hi ? 0.f : ay;

  // Row-side data for the D layout: this lane's 8 rows are M = (hi?8:0)+r
  const int rbase = rt * 16 + (hi ? 8 : 0);
  float rowc[8];                       // 0.5*sq_row - 0.5*R2
  unsigned bfr = 0;                    // row boundary bits (epilogue only)
  #pragma unroll
  for (int r = 0; r < 8; ++r) {
    float px = pts[(rbase + r) * 3 + 0];
    float py = pts[(rbase + r) * 3 + 1];
    float pz = pts[(rbase + r) * 3 + 2];
    rowc[r] = 0.5f * (px * px + py * py + pz * pz) - 0.5f * RAD2;
    if (bm[rbase + r]) bfr |= (1u << r);
  }

  float cnt[8], sx[8], sx2[8];
  #pragma unroll
  for (int r = 0; r < 8; ++r) { cnt[r] = 0.f; sx[r] = 0.f; sx2[r] = 0.f; }

  for (int ct = 0; ct < NPTS / 16; ++ct) {
    const int col = ct * 16 + l;               // this lane's column N = l
    const float qx = pts[col * 3 + 0];
    const float qy = pts[col * 3 + 1];
    const float qz = pts[col * 3 + 2];
    // B operand (4x16 f32), symmetric layout: lane->N, low half (K0,K1), high half (K2,K3)
    v2f B;
    B.x = hi ? qz : qx;
    B.y = hi ? 0.f : qy;
    const float xj = seg[col * 3 + 2];
    // column threshold: +inf disables the whole column when bf_j == 0
    const float hs = bm[col] ? 0.5f * (qx * qx + qy * qy + qz * qz)
                             : __builtin_inff();

    v8f C = {};
    v8f D;
#if __has_builtin(__builtin_amdgcn_wmma_f32_16x16x4_f32)
    D = __builtin_amdgcn_wmma_f32_16x16x4_f32(false, A, false, B, (short)0, C,
                                              false, false);
#else
    asm volatile("v_wmma_f32_16x16x4_f32 %0, %1, %2, %3"
                 : "=v"(D) : "v"(A), "v"(B), "v"(C));
#endif

    const float xj2 = xj * xj;
    #pragma unroll
    for (int r = 0; r < 8; ++r) {
      // D[r] = dot(p_row, p_col) at (M = (hi?8:0)+r, N = l)
      float m = (D[r] > rowc[r] + hs) ? 1.0f : 0.0f;
      cnt[r] += m;
      sx[r]   = fmaf(m, xj,  sx[r]);
      sx2[r]  = fmaf(m, xj2, sx2[r]);
    }
  }

  // Reduce across the 16 lanes of each half (xor masks 1,2,4,8 stay within half)
  #pragma unroll
  for (int r = 0; r < 8; ++r) {
    for (int m = 1; m < 16; m <<= 1) {
      cnt[r] += __shfl_xor(cnt[r], m, 32);
      sx[r]  += __shfl_xor(sx[r],  m, 32);
      sx2[r] += __shfl_xor(sx2[r], m, 32);
    }
  }

  if (l == 0) {  // lane 0 (rows 0..7) and lane 16 (rows 8..15)
    float lv = 0.f, lni = 0.f;
    #pragma unroll
    for (int r = 0; r < 8; ++r) {
      float c  = cnt[r];
      float sn = fmaxf(c, 2.0f);
      float var = (sx2[r] - sx[r] * sx[r] / sn) / (sn - 1.0f);
      if ((c > 1.0f) && ((bfr >> r) & 1u)) { lv += var; lni += 1.f; }
    }
    atomicAdd(&ws[16], lv);
    atomicAdd(&ws[17], lni);
  }
}

__global__ void k_final(const float* __restrict__ ws, float* __restrict__ out) {
  if (threadIdx.x != 0 || blockIdx.x != 0) return;
  const float pv = ws[0], gv = ws[1], nh = ws[2];
  const float sb = ws[6], Sd = ws[7], Sd2 = ws[8];
  const float maxd = __uint_as_float(((const unsigned*)ws)[9]);
  const float sum_var = ws[16], ni = ws[17];

  // --- ellipsoid shape loss (3x3 symmetric eigensolve, Cardano) ---
  float el = 0.f;
  if (nh >= 10.f) {
    const float inv = 1.f / nh;
    float a11 = ws[10] * inv, a12 = ws[11] * inv, a13 = ws[12] * inv;
    float a22 = ws[13] * inv, a23 = ws[14] * inv, a33 = ws[15] * inv;
    float p1 = a12 * a12 + a13 * a13 + a23 * a23;
    float w0, w1, w2;  // ascending
    if (p1 <= 0.f) {
      float lo  = fminf(a11, fminf(a22, a33));
      float hiv = fmaxf(a11, fmaxf(a22, a33));
      w0 = lo; w2 = hiv; w1 = a11 + a22 + a33 - lo - hiv;
    } else {
      float q  = (a11 + a22 + a33) * (1.f / 3.f);
      float p2 = (a11 - q) * (a11 - q) + (a22 - q) * (a22 - q) +
                 (a33 - q) * (a33 - q) + 2.f * p1;
      float p  = sqrtf(p2 * (1.f / 6.f));
      float ip = 1.f / p;
      float b11 = (a11 - q) * ip, b12 = a12 * ip, b13 = a13 * ip;
      float b22 = (a22 - q) * ip, b23 = a23 * ip, b33 = (a33 - q) * ip;
      float det = b11 * (b22 * b33 - b23 * b23) -
                  b12 * (b12 * b33 - b23 * b13) +
                  b13 * (b12 * b23 - b22 * b13);
      float r = fminf(1.f, fmaxf(-1.f, det * 0.5f));
      float phi = acosf(r) * (1.f / 3.f);
      float e1 = q + 2.f * p * cosf(phi);
      float e3 = q + 2.f * p * cosf(phi + 2.0943951023931953f);  // +2pi/3
      float e2 = 3.f * q - e1 - e3;
      w2 = e1; w1 = e2; w0 = e3;
    }
    float t1 = w1 / (w2 + EPSL) - 1.f;
    float t2 = w0 / (w2 + EPSL) - 1.f;
    el = t1 * t1 + t2 * t2;
  }

  // --- size consistency ---
  float diff = pv - gv;
  float sz = diff * diff;
  if (gv > 0.f) sz += 0.5f * fabsf(diff) / gv;

  // --- surface smoothness ---
  float sm = 0.f;
  if (sb >= 5.f && ni > 0.f) sm = sum_var / fmaxf(ni, 1.f);

  // --- connectivity ---
  float conn = 0.f;
  if (nh >= 5.f) {
    float mean = Sd / nh;
    float var  = (Sd2 - nh * mean * mean) / (nh - 1.f);
    conn = var / (maxd + EPSL);
  }

  out[0] = el + sz + sm + conn;
}

extern "C" void kernel_launch(void* const* d_in, const int* in_sizes, int n_in,
                              void* d_out, int out_size, void* d_ws, size_t ws_size,
                              hipStream_t stream) {
  const float* pts = (const float*)d_in[0];  // points (N,3)
  const float* seg = (const float*)d_in[1];  // seg_probs (N,3)
  const int* bm = (const int*)d_in[2];       // boundary_mask
  const int* hm = (const int*)d_in[3];       // head_mask
  const int* pm = (const int*)d_in[4];       // pred_mask
  const int* gm = (const int*)d_in[5];       // gt_mask
  float* ws  = (float*)d_ws;
  float* out = (float*)d_out;

  k_init<<<1, 32, 0, stream>>>(ws);
  k_reduce1<<<32, 256, 0, stream>>>(pts, hm, pm, gm, bm, ws);
  k_reduce2<<<32, 256, 0, stream>>>(pts, hm, ws);
  k_smooth<<<NPTS / 16 / 8, 256, 0, stream>>>(pts, seg, bm, ws);  // 64 blocks x 8 waves = 512 row tiles
  k_final<<<1, 1, 0, stream>>>(ws, out);
}